// SkeletonConvolution_45037027066536
// MI455X (gfx1250) — compile-verified
//
#include <hip/hip_runtime.h>
#include <hip/hip_bf16.h>

// Skeleton-sparse 1D conv as block GEMMs on CDNA5 WMMA (f32 16x16x4).
//
// out[b, co, t] = bias[co] + sum_{ci in nbrs, k} Wmask[co,ci,k] * x_reflect[b, ci, t+k-7]
//
// Joint-pair tiling: output joints (2p, 2p+1) -> M=16; neighbor-union -> 16
// input channels; GEMM K-dim = 240 ordered r = k*16 + ci (k-major) so each
// WMMA step's (ci, k) split is shifts/masks only. Fully unrolled K loop ->
// all LDS accesses are immediate-offset ds_load from 2 base pointers.
// Mask realized by zero-filling non-neighbor weights in LDS.
// Memory-bound: >= ~604 MB @ 23.3 TB/s => ~26 us floor; f32 WMMA keeps
// reference precision while the compute (~48 GFLOP) rides in the matrix pipe.

typedef __attribute__((ext_vector_type(2))) float v2f;
typedef __attribute__((ext_vector_type(8))) float v8f;

#define NJ      24
#define CIN_PJ  4
#define COUT_PJ 8
#define KSZ     15
#define PADW    7
#define BATCH   128
#define TLEN    4096
#define CIN     (NJ * CIN_PJ)   /* 96  */
#define COUT    (NJ * COUT_PJ)  /* 192 */

#define TCOLS   256                  /* time columns per block (4 waves x 64) */
#define XCOLS   (TCOLS + 2 * PADW)   /* 270 staged x columns                  */
#define XSTR    272                  /* padded LDS row stride (floats)        */
#define KDIM    (16 * KSZ)           /* 240 = GEMM K dimension                */

__global__ __launch_bounds__(128)
void skel_conv_wmma_f32(const float* __restrict__ x,
                        const float* __restrict__ weight,
                        const float* __restrict__ bias,
                        float* __restrict__ out)
{
    __shared__ float sW[KDIM * 16];   // [r = k*16+ci][m], masked weights
    __shared__ float sX[16 * XSTR];   // [ci][col] reflected x tile

    const int tid  = threadIdx.x;
    const int lane = tid & 31;
    const int wave = tid >> 5;
    const int m    = lane & 15;   // M (A/D) or N (B) index within 16
    const int hi   = lane >> 4;   // 0: K rows +0/+1, 1: K rows +2/+3
    const int koff = hi << 1;     // 0 or 2

    const int p    = blockIdx.x;  // 0..11  joint pair (fastest: x-slice L2 reuse)
    const int tile = blockIdx.y;  // 0..15  time tile
    const int b    = blockIdx.z;  // 0..127 batch

    const int t0 = tile * TCOLS;

    int jstart = 2 * p - 1;                   // first joint of 4-joint union
    if (jstart < 0)       jstart = 0;
    if (jstart > NJ - 4)  jstart = NJ - 4;
    const int cbase = jstart * CIN_PJ;

    // ---- stage masked weights: sW[(k*16+ci)*16 + m] = Wmask[p*16+m, cbase+ci, k]
    for (int idx = tid; idx < KDIM * 16; idx += 128) {
        const int r    = idx >> 4;            // r = k*16 + ci
        const int mloc = idx & 15;
        const int ci   = r & 15;
        const int k    = r >> 4;
        const int d    = (jstart + (ci >> 2)) - (2 * p + (mloc >> 3));
        float wv = 0.0f;
        if (d >= -1 && d <= 1) {              // exact skeleton-neighbour mask
            wv = weight[((p * 16 + mloc) * CIN + (cbase + ci)) * KSZ + k];
        }
        sW[idx] = wv;
    }

    // ---- stage x tile (16 ch x 270 cols) with reflect padding; no int div
    for (int ch = 0; ch < 16; ++ch) {
        const float* __restrict__ xrow =
            &x[((size_t)b * CIN + (cbase + ch)) * TLEN];
        float* __restrict__ srow = &sX[ch * XSTR];
        for (int col = tid; col < XCOLS; col += 128) {
            int u = t0 - PADW + col;
            if (u < 0)      u = -u;           // jnp 'reflect' (no edge repeat)
            if (u >= TLEN)  u = 2 * TLEN - 2 - u;
            srow[col] = xrow[u];
        }
    }

    __syncthreads();

    // Each wave owns 64 time columns = four 16-wide WMMA column tiles.
    const int nb = wave * 64;

    v8f acc0 = {}, acc1 = {}, acc2 = {}, acc3 = {};

    // Base pointers; all per-step offsets are compile-time immediates.
    const float* __restrict__ pA = &sW[koff * 16 + m];
    const float* __restrict__ pB = &sX[koff * XSTR + m + nb];

#pragma unroll
    for (int kk = 0; kk < KDIM; kk += 4) {
        const int aoff = kk * 16;                       // r0*16 (minus base)
        const int boff = (kk & 15) * XSTR + (kk >> 4);  // ci0*XSTR + k

        v2f a;                       // A 16x4: V0 = K{0|2}, V1 = K{1|3}
        a.x = pA[aoff];
        a.y = pA[aoff + 16];

        v2f b0, b1, b2, b3;          // B 4x16: same K striping, N = lane%16
        b0.x = pB[boff];        b0.y = pB[boff + XSTR];
        b1.x = pB[boff + 16];   b1.y = pB[boff + XSTR + 16];
        b2.x = pB[boff + 32];   b2.y = pB[boff + XSTR + 32];
        b3.x = pB[boff + 48];   b3.y = pB[boff + XSTR + 48];

        acc0 = __builtin_amdgcn_wmma_f32_16x16x4_f32(
            false, a, false, b0, (short)0, acc0, false, false);
        acc1 = __builtin_amdgcn_wmma_f32_16x16x4_f32(
            false, a, false, b1, (short)0, acc1, false, false);
        acc2 = __builtin_amdgcn_wmma_f32_16x16x4_f32(
            false, a, false, b2, (short)0, acc2, false, false);
        acc3 = __builtin_amdgcn_wmma_f32_16x16x4_f32(
            false, a, false, b3, (short)0, acc3, false, false);
    }

    // ---- epilogue: D layout (VGPR v -> M = v + 8*hi, N = lane%16), add bias
#pragma unroll
    for (int v = 0; v < 8; ++v) {
        const int co = p * 16 + hi * 8 + v;
        const float bz = bias[co];
        float* __restrict__ orow =
            &out[((size_t)b * COUT + co) * TLEN + t0 + nb + m];
        orow[0]  = acc0[v] + bz;
        orow[16] = acc1[v] + bz;
        orow[32] = acc2[v] + bz;
        orow[48] = acc3[v] + bz;
    }
}

extern "C" void kernel_launch(void* const* d_in, const int* in_sizes, int n_in,
                              void* d_out, int out_size, void* d_ws, size_t ws_size,
                              hipStream_t stream) {
    const float* x      = (const float*)d_in[0];   // [128, 96, 4096]
    const float* weight = (const float*)d_in[1];   // [192, 96, 15]
    const float* bias   = (const float*)d_in[2];   // [192]
    float*       out    = (float*)d_out;           // [128, 192, 4096]

    dim3 grid(NJ / 2, TLEN / TCOLS, BATCH);        // (12, 16, 128)
    dim3 block(128);                               // 4 wave32
    skel_conv_wmma_f32<<<grid, block, 0, stream>>>(x, weight, bias, out);
}